// GAT_82205674045442
// MI455X (gfx1250) — compile-verified
//
#include <hip/hip_runtime.h>
#include <hip/hip_bf16.h>
#include <math.h>

// ---- CDNA5 WMMA types -------------------------------------------------
typedef float v2f __attribute__((ext_vector_type(2)));
typedef float v8f __attribute__((ext_vector_type(8)));

#define NEG_SLOPE 0.2f
#define BN_EPS    1e-5f

// Order-preserving float<->uint encoding for atomic segment-max.
__device__ __forceinline__ unsigned fenc(float f) {
  unsigned u = __float_as_uint(f);
  return (u & 0x80000000u) ? ~u : (u | 0x80000000u);
}
__device__ __forceinline__ float fdec(unsigned u) {
  unsigned v = (u & 0x80000000u) ? (u & 0x7FFFFFFFu) : ~u;
  return __uint_as_float(v);
}
__device__ __forceinline__ float lrelu(float x) { return x > 0.f ? x : NEG_SLOPE * x; }

// ---- init accumulators ------------------------------------------------
__global__ void k_init(float* out1, float* out2, float* den1, unsigned* m1,
                       float* den2, unsigned* m2, int n) {
  int i = blockIdx.x * blockDim.x + threadIdx.x;
  if (i < n * 128) out1[i] = 0.f;
  if (i < n * 40)  out2[i] = 0.f;
  if (i < n * 8)  { den1[i] = 0.f; m1[i] = 0x007FFFFFu; }  // fenc(-inf)
  if (i < n)      { den2[i] = 0.f; m2[i] = 0x007FFFFFu; }
}

// ---- fp32 WMMA GEMM: C[n x M] = A[n x 128] * B[128 x M] (+bias) -------
// Block = 8 waves: one 16-col panel of B staged in LDS, 8 row tiles.
// Per wave: 16x16 output tile via 32 x v_wmma_f32_16x16x4_f32.
// LDS layout Bs[col*132 + k]: fragment read = conflict-free ds_load_b64.
template <int M>
__global__ __launch_bounds__(256)
void k_gemm_wmma(const float* __restrict__ A, const float* __restrict__ B,
                 const float* __restrict__ bias, float* __restrict__ C, int n) {
  constexpr int CT = (M + 15) / 16;          // column tiles
  __shared__ float Bs[16 * 132];

  int colTile  = (int)(blockIdx.x) % CT;
  int rowBlock = (int)(blockIdx.x) / CT;
  int col0 = colTile * 16;

  // Cooperative fill of the B panel (128 K x 16 cols), zero-padded past M.
#pragma unroll
  for (int i = 0; i < 8; ++i) {
    int idx = (int)threadIdx.x + i * 256;    // 0..2047
    int k   = idx & 127;
    int c   = idx >> 7;                      // 0..15
    int gc  = col0 + c;
    Bs[c * 132 + k] = (gc < M) ? B[(size_t)k * M + gc] : 0.f;
  }
  __syncthreads();

  int wv   = threadIdx.x >> 5;
  int lane = threadIdx.x & 31;
  int row0 = (rowBlock * 8 + wv) * 16;
  if (row0 >= n) return;                     // n % 16 == 0; wave-uniform

  int half = lane >> 4;                      // 0: K+{0,1}, 1: K+{2,3}
  int koff = half * 2;
  int l16  = lane & 15;
  const float* Arow = A + (size_t)(row0 + l16) * 128;
  const float* Bcol = &Bs[l16 * 132 + koff];

  v8f acc = {0.f, 0.f, 0.f, 0.f, 0.f, 0.f, 0.f, 0.f};
#pragma unroll
  for (int k0 = 0; k0 < 128; k0 += 4) {
    v2f a = *(const v2f*)(Arow + k0 + koff); // global_load_b64
    v2f b = *(const v2f*)(Bcol + k0);        // ds_load_b64, conflict-free
    acc = __builtin_amdgcn_wmma_f32_16x16x4_f32(false, a, false, b,
                                                (short)0, acc, false, false);
  }

  int col = col0 + l16;
  if (col < M) {
    float bv = bias ? bias[col] : 0.f;
    int rbase = row0 + half * 8;             // acc[v]: row v (lanes 0-15) / v+8
#pragma unroll
    for (int v = 0; v < 8; ++v)
      C[(size_t)(rbase + v) * M + col] = acc[v] + bv;
  }
}

// ---- per-node attention dots, layer 1: alpha[n,h] = sum_c h1[n,h,c]*att[h,c]
__global__ void k_alpha1(const float* __restrict__ h1, const float* __restrict__ asrc,
                         const float* __restrict__ adst, float* as1, float* ad1, int n) {
  int i = blockIdx.x * blockDim.x + threadIdx.x;
  if (i >= n * 8) return;
  int h = i & 7;
  const float* hp = h1 + (size_t)(i >> 3) * 128 + h * 16;
  const float* s = asrc + h * 16;
  const float* d = adst + h * 16;
  float ss = 0.f, dd = 0.f;
#pragma unroll
  for (int c = 0; c < 16; ++c) { float v = hp[c]; ss += v * s[c]; dd += v * d[c]; }
  as1[i] = ss; ad1[i] = dd;
}

// ---- layer-1 edge passes (one thread per (edge, head)) ----------------
__device__ __forceinline__ void edge_sd(const int* ei, int E, int e, int& s, int& d) {
  if (e < E) { s = ei[e]; d = ei[E + e]; } else { s = d = e - E; }  // self-loops
}

__global__ void k_e1_max(const int* __restrict__ ei, int E, int n,
                         const float* __restrict__ as1, const float* __restrict__ ad1,
                         unsigned* m1) {
  int i = blockIdx.x * blockDim.x + threadIdx.x;
  if (i >= (E + n) * 8) return;
  int h = i & 7, e = i >> 3, s, d;
  edge_sd(ei, E, e, s, d);
  float ev = lrelu(as1[s * 8 + h] + ad1[d * 8 + h]);
  atomicMax(&m1[d * 8 + h], fenc(ev));
}

__global__ void k_e1_sum(const int* __restrict__ ei, int E, int n,
                         const float* __restrict__ as1, const float* __restrict__ ad1,
                         const unsigned* __restrict__ m1, float* den1) {
  int i = blockIdx.x * blockDim.x + threadIdx.x;
  if (i >= (E + n) * 8) return;
  int h = i & 7, e = i >> 3, s, d;
  edge_sd(ei, E, e, s, d);
  int idx = d * 8 + h;
  float ev = lrelu(as1[s * 8 + h] + ad1[idx]);
  atomicAdd(&den1[idx], expf(ev - fdec(m1[idx])));
}

__global__ void k_e1_agg(const int* __restrict__ ei, int E, int n,
                         const float* __restrict__ as1, const float* __restrict__ ad1,
                         const unsigned* __restrict__ m1, const float* __restrict__ den1,
                         const float* __restrict__ h1, float* out1) {
  int i = blockIdx.x * blockDim.x + threadIdx.x;
  if (i >= (E + n) * 8) return;
  int h = i & 7, e = i >> 3, s, d;
  edge_sd(ei, E, e, s, d);
  int idx = d * 8 + h;
  float ev = lrelu(as1[s * 8 + h] + ad1[idx]);
  float alpha = expf(ev - fdec(m1[idx])) / (den1[idx] + 1e-16f);
  const float* hp = h1 + (size_t)s * 128 + h * 16;
  float* op = out1 + (size_t)d * 128 + h * 16;
#pragma unroll
  for (int c = 0; c < 16; ++c) atomicAdd(&op[c], alpha * hp[c]);
}

// ---- fused bias1 + BN(eval) + ELU + skip ------------------------------
__global__ void k_bn_elu_skip(const float* __restrict__ out1, const float* __restrict__ ident,
                              const float* __restrict__ b1, const float* __restrict__ g,
                              const float* __restrict__ be, const float* __restrict__ mu,
                              const float* __restrict__ var, float* h2, int n) {
  int i = blockIdx.x * blockDim.x + threadIdx.x;
  if (i >= n * 128) return;
  int c = i & 127;
  float v = out1[i] + b1[c];
  v = (v - mu[c]) * (g[c] * rsqrtf(var[c] + BN_EPS)) + be[c];
  v = v > 0.f ? v : (expf(v) - 1.f);   // ELU, alpha=1
  h2[i] = v + ident[i];
}

// ---- per-node attention dots, layer 2 (heads=1, C=40) -----------------
__global__ void k_alpha2(const float* __restrict__ g2, const float* __restrict__ asrc,
                         const float* __restrict__ adst, float* as2, float* ad2, int n) {
  int i = blockIdx.x * blockDim.x + threadIdx.x;
  if (i >= n) return;
  const float* p = g2 + (size_t)i * 40;
  float ss = 0.f, dd = 0.f;
#pragma unroll
  for (int c = 0; c < 40; ++c) { float v = p[c]; ss += v * asrc[c]; dd += v * adst[c]; }
  as2[i] = ss; ad2[i] = dd;
}

// ---- layer-2 edge passes (one thread per edge) ------------------------
__global__ void k_e2_max(const int* __restrict__ ei, int E, int n,
                         const float* __restrict__ as2, const float* __restrict__ ad2,
                         unsigned* m2) {
  int e = blockIdx.x * blockDim.x + threadIdx.x;
  if (e >= E + n) return;
  int s, d; edge_sd(ei, E, e, s, d);
  atomicMax(&m2[d], fenc(lrelu(as2[s] + ad2[d])));
}

__global__ void k_e2_sum(const int* __restrict__ ei, int E, int n,
                         const float* __restrict__ as2, const float* __restrict__ ad2,
                         const unsigned* __restrict__ m2, float* den2) {
  int e = blockIdx.x * blockDim.x + threadIdx.x;
  if (e >= E + n) return;
  int s, d; edge_sd(ei, E, e, s, d);
  float ev = lrelu(as2[s] + ad2[d]);
  atomicAdd(&den2[d], expf(ev - fdec(m2[d])));
}

__global__ void k_e2_agg(const int* __restrict__ ei, int E, int n,
                         const float* __restrict__ as2, const float* __restrict__ ad2,
                         const unsigned* __restrict__ m2, const float* __restrict__ den2,
                         const float* __restrict__ g2, float* out2) {
  int e = blockIdx.x * blockDim.x + threadIdx.x;
  if (e >= E + n) return;
  int s, d; edge_sd(ei, E, e, s, d);
  float ev = lrelu(as2[s] + ad2[d]);
  float alpha = expf(ev - fdec(m2[d])) / (den2[d] + 1e-16f);
  const float* gp = g2 + (size_t)s * 40;
  float* op = out2 + (size_t)d * 40;
#pragma unroll
  for (int c = 0; c < 40; ++c) atomicAdd(&op[c], alpha * gp[c]);
}

// ---- bias2 + log_softmax ---------------------------------------------
__global__ void k_logsoftmax(const float* __restrict__ out2, const float* __restrict__ b2,
                             float* __restrict__ y, int n) {
  int i = blockIdx.x * blockDim.x + threadIdx.x;
  if (i >= n) return;
  const float* p = out2 + (size_t)i * 40;
  float v[40];
  float mx = __uint_as_float(0xFF800000u);  // -inf
#pragma unroll
  for (int c = 0; c < 40; ++c) { v[c] = p[c] + b2[c]; mx = fmaxf(mx, v[c]); }
  float sum = 0.f;
#pragma unroll
  for (int c = 0; c < 40; ++c) sum += expf(v[c] - mx);
  float ls = mx + logf(sum);
  float* o = y + (size_t)i * 40;
#pragma unroll
  for (int c = 0; c < 40; ++c) o[c] = v[c] - ls;
}

// ======================================================================
extern "C" void kernel_launch(void* const* d_in, const int* in_sizes, int n_in,
                              void* d_out, int out_size, void* d_ws, size_t ws_size,
                              hipStream_t stream) {
  (void)n_in; (void)out_size; (void)ws_size;
  const float* x    = (const float*)d_in[0];
  const int*   ei   = (const int*)  d_in[1];
  const float* W1   = (const float*)d_in[2];
  const float* at_s1= (const float*)d_in[3];
  const float* at_d1= (const float*)d_in[4];
  const float* b1   = (const float*)d_in[5];
  const float* bng  = (const float*)d_in[6];
  const float* bnb  = (const float*)d_in[7];
  const float* bnm  = (const float*)d_in[8];
  const float* bnv  = (const float*)d_in[9];
  const float* W2   = (const float*)d_in[10];
  const float* at_s2= (const float*)d_in[11];
  const float* at_d2= (const float*)d_in[12];
  const float* b2   = (const float*)d_in[13];
  const float* Wsk  = (const float*)d_in[14];
  const float* bsk  = (const float*)d_in[15];

  int n = in_sizes[0] / 128;   // 100000
  int E = in_sizes[1] / 2;     // 1600000

  // workspace carve-out (floats)
  float* ws = (float*)d_ws;
  size_t o = 0;
  float*    h1   = ws + o; o += (size_t)n * 128;   // x@W1 (later reused for h2in)
  float*    idnt = ws + o; o += (size_t)n * 128;   // x@W_skip + b_skip
  float*    out1 = ws + o; o += (size_t)n * 128;   // layer-1 aggregation
  float*    g2   = ws + o; o += (size_t)n * 40;    // h2@W2
  float*    out2 = ws + o; o += (size_t)n * 40;    // layer-2 aggregation
  float*    as1  = ws + o; o += (size_t)n * 8;
  float*    ad1  = ws + o; o += (size_t)n * 8;
  float*    den1 = ws + o; o += (size_t)n * 8;
  unsigned* m1   = (unsigned*)(ws + o); o += (size_t)n * 8;
  float*    as2  = ws + o; o += (size_t)n;
  float*    ad2  = ws + o; o += (size_t)n;
  float*    den2 = ws + o; o += (size_t)n;
  unsigned* m2   = (unsigned*)(ws + o); o += (size_t)n;

  const int T = 256;
  auto cdiv = [](long a, long b) { return (unsigned)((a + b - 1) / b); };

  k_init<<<cdiv((long)n * 128, T), T, 0, stream>>>(out1, out2, den1, m1, den2, m2, n);

  int rowTiles  = (n + 15) / 16;               // 6250
  int rowBlocks = (rowTiles + 7) / 8;          // 8 row tiles per block

  // h1 = x @ W1 (no bias; bias1 folded in later)
  k_gemm_wmma<128><<<8 * rowBlocks, 256, 0, stream>>>(x, W1, nullptr, h1, n);
  // identity = x @ W_skip + b_skip
  k_gemm_wmma<128><<<8 * rowBlocks, 256, 0, stream>>>(x, Wsk, bsk, idnt, n);

  k_alpha1<<<cdiv((long)n * 8, T), T, 0, stream>>>(h1, at_s1, at_d1, as1, ad1, n);

  long et1 = (long)(E + n) * 8;
  k_e1_max<<<cdiv(et1, T), T, 0, stream>>>(ei, E, n, as1, ad1, m1);
  k_e1_sum<<<cdiv(et1, T), T, 0, stream>>>(ei, E, n, as1, ad1, m1, den1);
  k_e1_agg<<<cdiv(et1, T), T, 0, stream>>>(ei, E, n, as1, ad1, m1, den1, h1, out1);

  // h2 = elu(bn(out1 + b1)) + identity  (written over h1 buffer)
  k_bn_elu_skip<<<cdiv((long)n * 128, T), T, 0, stream>>>(out1, idnt, b1, bng, bnb, bnm, bnv, h1, n);

  // g2 = h2 @ W2   (M = 40 -> 3 col tiles, zero-padded in LDS)
  k_gemm_wmma<40><<<3 * rowBlocks, 256, 0, stream>>>(h1, W2, nullptr, g2, n);

  k_alpha2<<<cdiv(n, T), T, 0, stream>>>(g2, at_s2, at_d2, as2, ad2, n);

  long et2 = E + n;
  k_e2_max<<<cdiv(et2, T), T, 0, stream>>>(ei, E, n, as2, ad2, m2);
  k_e2_sum<<<cdiv(et2, T), T, 0, stream>>>(ei, E, n, as2, ad2, m2, den2);
  k_e2_agg<<<cdiv(et2, T), T, 0, stream>>>(ei, E, n, as2, ad2, m2, den2, g2, out2);

  k_logsoftmax<<<cdiv(n, T), T, 0, stream>>>(out2, b2, (float*)d_out, n);
}